// EventCategorizationHead_11424613007667
// MI455X (gfx1250) — compile-verified
//
#include <hip/hip_runtime.h>
#include <math.h>

typedef __attribute__((ext_vector_type(2))) float v2f;
typedef __attribute__((ext_vector_type(4))) float v4f;
typedef __attribute__((ext_vector_type(8))) float v8f;

#define LN_EPS 1e-5f
#define IN_CH 256
#define PTS_PER_BLK 1024

// ---------------- zero the accumulator ----------------
__global__ void zero_kernel(float* __restrict__ p, int n) {
  int i = blockIdx.x * blockDim.x + threadIdx.x;
  if (i < n) p[i] = 0.0f;
}

// ---------------- Stage 1: segment sum (bandwidth bound) ----------------
// feat: [npts, 256] row-major; offsets: [nseg+1]; sums: [nseg, 256]
__device__ __forceinline__ void flush_acc(float* __restrict__ sums, int seg,
                                          int chb, float4& a) {
  if (a.x != 0.f || a.y != 0.f || a.z != 0.f || a.w != 0.f) {
    atomicAdd(&sums[seg * IN_CH + chb + 0], a.x);
    atomicAdd(&sums[seg * IN_CH + chb + 1], a.y);
    atomicAdd(&sums[seg * IN_CH + chb + 2], a.z);
    atomicAdd(&sums[seg * IN_CH + chb + 3], a.w);
  }
  a = make_float4(0.f, 0.f, 0.f, 0.f);
}

__global__ void segsum_kernel(const float* __restrict__ feat,
                              const int* __restrict__ offsets,
                              float* __restrict__ sums,
                              int npts, int nseg) {
  __shared__ int offs[1024];
  int tid = threadIdx.x;
  for (int i = tid; i <= nseg; i += blockDim.x) offs[i] = offsets[i];
  __syncthreads();

  int p0 = blockIdx.x * PTS_PER_BLK;
  int p1 = min(p0 + PTS_PER_BLK, npts);
  int quad = tid >> 6;            // which of 4 points per iteration
  int chb  = (tid & 63) * 4;      // channel base (float4 per thread)
  int p = p0 + quad;
  if (p >= p1) return;

  // largest seg with offs[seg] <= p  (offs sorted; offs[nseg] == npts > p)
  int lo = 0, hi = nseg - 1;
  while (lo < hi) {
    int mid = (lo + hi + 1) >> 1;
    if (offs[mid] <= p) lo = mid; else hi = mid - 1;
  }
  int seg = lo;
  int segEnd = offs[seg + 1];

  float4 acc = make_float4(0.f, 0.f, 0.f, 0.f);
  for (; p < p1; p += 4) {
    while (p >= segEnd) {             // crossing (possibly empty) segments
      flush_acc(sums, seg, chb, acc);
      ++seg;
      segEnd = offs[seg + 1];
    }
    const v4f* src = (const v4f*)(feat + (long)p * IN_CH + chb);
    v4f v = __builtin_nontemporal_load(src);   // streaming: no L2 reuse
    acc.x += v.x; acc.y += v.y; acc.z += v.z; acc.w += v.w;
  }
  flush_acc(sums, seg, chb, acc);
}

// ---------------- Stage 2: divide by counts (in place) ----------------
__global__ void mean_kernel(float* __restrict__ sums,
                            const int* __restrict__ offsets) {
  int b = blockIdx.x, c = threadIdx.x;
  int cnt = offsets[b + 1] - offsets[b];
  float inv = 1.0f / (float)max(cnt, 1);
  sums[b * IN_CH + c] *= inv;
}

// ---------------- f32 WMMA GEMM: out[M,N] = A[M,K] @ W[K,N] + bias --------
// One wave (32 lanes) per 16x16 output tile; V_WMMA_F32_16X16X4_F32, K step 4.
// A lane layout (16x4 f32): lanes 0-15 -> K=0,1 ; lanes 16-31 -> K=2,3.
// C/D: 8 VGPRs; VGPR r, lanes 0-15 -> M=r, lanes 16-31 -> M=r+8; N = lane%16.
// UNGUARDED: N must be a multiple of 16 and all loads in bounds (pad first).
__global__ void gemm16_wmma(const float* __restrict__ A,
                            const float* __restrict__ W,
                            const float* __restrict__ bias,
                            float* __restrict__ out,
                            int K, int N) {
  int lane = threadIdx.x & 31;
  int half = lane >> 4;           // 0 or 1
  int l    = lane & 15;
  int row  = blockIdx.y * 16 + l; // A row this lane feeds
  int col  = blockIdx.x * 16 + l; // B/out column this lane feeds

  const float* Arow = A + (size_t)row * K;
  const float* Wcol = W + col;

  v8f c = {};
  for (int k0 = 0; k0 < K; k0 += 4) {
    int ka = k0 + half * 2;
    v2f a, b;
    a.x = Arow[ka];
    a.y = Arow[ka + 1];
    b.x = Wcol[(size_t)ka * N];
    b.y = Wcol[(size_t)(ka + 1) * N];
    c = __builtin_amdgcn_wmma_f32_16x16x4_f32(
        /*neg_a=*/false, a, /*neg_b=*/false, b,
        /*c_mod=*/(short)0, c, /*reuse_a=*/false, /*reuse_b=*/false);
  }
  float bb = bias[col];
#pragma unroll
  for (int r = 0; r < 8; ++r) {
    int m = blockIdx.y * 16 + r + half * 8;
    out[(size_t)m * N + col] = c[r] + bb;
  }
}

// ---------------- LayerNorm + exact GELU (in place), one block per row ----
template <int H>
__global__ void ln_gelu_kernel(float* __restrict__ x,
                               const float* __restrict__ g,
                               const float* __restrict__ be) {
  constexpr int T = 256;
  constexpr int E = H / T;
  __shared__ float s1[T], s2[T];
  int row = blockIdx.x, tid = threadIdx.x;

  float v[E];
  float sum = 0.f, sq = 0.f;
#pragma unroll
  for (int e = 0; e < E; ++e) {
    v[e] = x[row * H + e * T + tid];
    sum += v[e];
    sq  += v[e] * v[e];
  }
  s1[tid] = sum; s2[tid] = sq;
  __syncthreads();
#pragma unroll
  for (int s = T / 2; s > 0; s >>= 1) {
    if (tid < s) { s1[tid] += s1[tid + s]; s2[tid] += s2[tid + s]; }
    __syncthreads();
  }
  float mu   = s1[0] / (float)H;
  float var  = s2[0] / (float)H - mu * mu;
  float rstd = rsqrtf(var + LN_EPS);
#pragma unroll
  for (int e = 0; e < E; ++e) {
    int cidx = e * T + tid;
    float y = (v[e] - mu) * rstd * g[cidx] + be[cidx];
    // exact GELU: 0.5*y*(1+erf(y/sqrt(2)))
    x[row * H + cidx] = 0.5f * y * (1.0f + erff(y * 0.70710678118654752f));
  }
}

// ---------------- pad W3 [K,10] -> W3p [K,16], b3 -> b3p[16] ----------------
__global__ void pad_w3_kernel(const float* __restrict__ W3,
                              const float* __restrict__ b3,
                              float* __restrict__ W3p,
                              float* __restrict__ b3p,
                              int K, int N, int Np) {
  int idx = blockIdx.x * blockDim.x + threadIdx.x;
  if (idx < K * Np) {
    int k = idx / Np, j = idx % Np;
    W3p[idx] = (j < N) ? W3[k * N + j] : 0.f;
  }
  if (idx < Np) b3p[idx] = (idx < N) ? b3[idx] : 0.f;
}

// ---------------- copy padded out [M,16] -> out [M,10] ----------------
__global__ void unpad_out_kernel(const float* __restrict__ outp,
                                 float* __restrict__ out,
                                 int M, int N, int Np) {
  int idx = blockIdx.x * blockDim.x + threadIdx.x;
  if (idx < M * N) {
    int m = idx / N, j = idx % N;
    out[idx] = outp[m * Np + j];
  }
}

extern "C" void kernel_launch(void* const* d_in, const int* in_sizes, int n_in,
                              void* d_out, int out_size, void* d_ws, size_t ws_size,
                              hipStream_t stream) {
  const float* feat    = (const float*)d_in[0];
  const int*   offsets = (const int*)d_in[1];
  const float* W1 = (const float*)d_in[2];
  const float* b1 = (const float*)d_in[3];
  const float* g1 = (const float*)d_in[4];
  const float* be1 = (const float*)d_in[5];
  const float* W2 = (const float*)d_in[6];
  const float* b2 = (const float*)d_in[7];
  const float* g2 = (const float*)d_in[8];
  const float* be2 = (const float*)d_in[9];
  const float* W3 = (const float*)d_in[10];
  const float* b3 = (const float*)d_in[11];
  float* out = (float*)d_out;

  int npts = in_sizes[0] / IN_CH;   // 1,000,000
  int nseg = in_sizes[1] - 1;       // 128

  // workspace layout (all 16B-aligned)
  char* ws = (char*)d_ws;
  size_t off = 0;
  float* ef   = (float*)(ws + off); off += (size_t)nseg * 256 * sizeof(float); // [nseg,256]
  float* h1   = (float*)(ws + off); off += (size_t)nseg * 512 * sizeof(float); // [nseg,512]
  float* h2   = (float*)(ws + off); off += (size_t)nseg * 256 * sizeof(float); // [nseg,256]
  float* W3p  = (float*)(ws + off); off += (size_t)256 * 16 * sizeof(float);   // [256,16]
  float* b3p  = (float*)(ws + off); off += (size_t)16 * sizeof(float);         // [16]
  float* outp = (float*)(ws + off); off += (size_t)nseg * 16 * sizeof(float);  // [nseg,16]

  zero_kernel<<<(nseg * IN_CH + 255) / 256, 256, 0, stream>>>(ef, nseg * IN_CH);

  int nblk = (npts + PTS_PER_BLK - 1) / PTS_PER_BLK;
  segsum_kernel<<<nblk, 256, 0, stream>>>(feat, offsets, ef, npts, nseg);

  mean_kernel<<<nseg, IN_CH, 0, stream>>>(ef, offsets);

  // pad classifier weights 256x10 -> 256x16 (runs concurrently-ish, tiny)
  pad_w3_kernel<<<(256 * 16 + 255) / 256, 256, 0, stream>>>(W3, b3, W3p, b3p,
                                                            256, 10, 16);

  // MLP: 256 -> 512 -> 256 -> 16(pad of 10), LN+GELU between
  gemm16_wmma<<<dim3(512 / 16, nseg / 16), 32, 0, stream>>>(ef, W1, b1, h1, 256, 512);
  ln_gelu_kernel<512><<<nseg, 256, 0, stream>>>(h1, g1, be1);
  gemm16_wmma<<<dim3(256 / 16, nseg / 16), 32, 0, stream>>>(h1, W2, b2, h2, 512, 256);
  ln_gelu_kernel<256><<<nseg, 256, 0, stream>>>(h2, g2, be2);
  gemm16_wmma<<<dim3(1, nseg / 16), 32, 0, stream>>>(h2, W3p, b3p, outp, 256, 16);

  unpad_out_kernel<<<(nseg * 10 + 255) / 256, 256, 0, stream>>>(outp, out,
                                                                nseg, 10, 16);
}